// SoftMatch_79018808312236
// MI455X (gfx1250) — compile-verified
//
#include <hip/hip_runtime.h>

typedef __attribute__((ext_vector_type(2))) float v2f;
typedef __attribute__((ext_vector_type(8))) float v8f;
typedef unsigned int u32;
typedef __attribute__((ext_vector_type(4))) u32 u32x4;
typedef __attribute__((ext_vector_type(8))) int i32x8;
typedef __attribute__((ext_vector_type(4))) int i32x4;

#define BB    1024      // batch rows
#define HH    768       // hidden
#define NN    100000    // memory rows
#define CC    42        // classes
#define KT    32        // K-tile staged per iteration
#define NTILES (HH / KT)
#define NCHUNK 64       // memory rows per workgroup
#define EPSF  1e-8f

#define HAVE_TDM __has_builtin(__builtin_amdgcn_tensor_load_to_lds)

// Generic LDS pointer -> byte offset within the workgroup LDS allocation.
// Per CDNA5 aperture mapping, the LDS offset is the low 32 address bits.
static __device__ __forceinline__ unsigned lds_byte_off(const void* p) {
    return (unsigned)(unsigned long long)p;
}

#if HAVE_TDM
// Issue one TDM 2-D tile load: dn[:, kt*KT : kt*KT+KT] (1024 rows x 32 f32,
// row stride 768 f32) compacted row-major into LDS at dstbuf. Wave-level op.
static __device__ __forceinline__ void tdm_dn_tile(const float* dn, int kt,
                                                   float* dstbuf) {
    const unsigned long long ga =
        (unsigned long long)dn + (unsigned long long)kt * (KT * sizeof(float));
    u32x4 g0;
    g0.x = 1u;                                   // count=1, user mode, no gather
    g0.y = lds_byte_off(dstbuf);                 // lds_addr (bytes)
    g0.z = (u32)ga;                              // global_addr[31:0]
    g0.w = (u32)((ga >> 32) & 0x01ffffffu)       // global_addr[56:32]
           | 0x80000000u;                        // type=2 ("image")
    i32x8 g1;
    g1[0] = 0x00020000;                          // data_size=4B; mask=0; no flags
    g1[1] = (HH & 0xffff) << 16;                 // tensor_dim0 = 768 (bits 63:48)
    g1[2] = (BB & 0xffff) << 16;                 // tensor_dim1 = 1024 (bits 95:80)
    g1[3] = KT << 16;                            // tile_dim0 = 32 (bits 127:112)
    g1[4] = BB;                                  // tile_dim1 = 1024 (bits 143:128)
    g1[5] = HH;                                  // tensor_dim0_stride = 768
    g1[6] = 0;
    g1[7] = 0;                                   // tensor_dim1_stride unused (2-D)
    const i32x4 gz4 = {0, 0, 0, 0};              // groups 2/3: unused (2-D tensor)
    const i32x8 gz8 = {0, 0, 0, 0, 0, 0, 0, 0};  // 6-arg toolchain extra group
    __builtin_amdgcn_tensor_load_to_lds(g0, g1, gz4, gz4, gz8, 0);
}
#endif

// ---------------------------------------------------------------------------
// Kernel 1: row-normalize weak_data -> dn (eps-clamped, faithful to reference)
// ---------------------------------------------------------------------------
__global__ __launch_bounds__(256) void normalize_kernel(
    const float* __restrict__ x, float* __restrict__ y)
{
    __shared__ float red[256];
    const int b = blockIdx.x;
    const int t = threadIdx.x;
    const float* row = x + (long)b * HH;
    float s = 0.f;
    for (int i = t; i < HH; i += 256) { float v = row[i]; s += v * v; }
    red[t] = s;
    __syncthreads();
    for (int w = 128; w > 0; w >>= 1) {
        if (t < w) red[t] += red[t + w];
        __syncthreads();
    }
    const float sc = 1.0f / fmaxf(sqrtf(red[0]), EPSF);
    float* out = y + (long)b * HH;
    for (int i = t; i < HH; i += 256) out[i] = row[i] * sc;
}

// ---------------------------------------------------------------------------
// Kernel 2: per-chunk GEMM (f32 WMMA) + fused memory-row norms + local argmax
//   grid.x = ceil(N/64); block = 256 (8 waves)
//   wave id w: g = w/4 (N-group of 32 cols), mg = w%4 (M-group of 256 rows)
//   dn K-panels double-buffered via TDM; B panels double-buffered via VGPRs.
// ---------------------------------------------------------------------------
__global__ __launch_bounds__(256) void gemm_argmax_kernel(
    const float* __restrict__ dn,    // [BB x HH] normalized weak rows
    const float* __restrict__ mem,   // [NN x HH] labeled memory (unnormalized)
    float* __restrict__ pval,        // [nchunks x BB] partial best value
    int*   __restrict__ pidx)        // [nchunks x BB] partial best index
{
    extern __shared__ float smem[];
    // layout: dnbuf0 | dnbuf1 | bbuf0 | bbuf1 | invn | lval0 | lidx0 | lval1 | lidx1
    float* dn_base = smem;                                 // 2 x BB*KT
    float* b_base  = smem + 2 * BB * KT;                   // 2 x NCHUNK*KT
    float* invn    = smem + 2 * BB * KT + 2 * NCHUNK * KT; // 64 f
    float* lval0   = invn + NCHUNK;                        // BB
    int*   lidx0   = (int*)(lval0 + BB);                   // BB
    float* lval1   = (float*)(lidx0 + BB);                 // BB
    int*   lidx1   = (int*)(lval1 + BB);                   // BB

    const int t    = threadIdx.x;
    const int wg   = blockIdx.x;
    const int wid  = t >> 5;
    const int lane = t & 31;
    const int g    = wid >> 2;      // N-group 0/1
    const int mg   = wid & 3;       // M-group 0..3
    const int lg   = lane >> 4;     // half-wave
    const int ln   = lane & 15;
    const int nbase = wg * NCHUNK;

    v8f acc[16][2];
#pragma unroll
    for (int m = 0; m < 16; ++m) { acc[m][0] = (v8f){}; acc[m][1] = (v8f){}; }

    // staging ownership for the B panel / norm accumulation
    const int brow = t >> 2;                 // 0..63 memory row within chunk
    const int bq0  = (t & 3) * 2;            // float4 slot pair within K-tile
    const int gn   = nbase + brow;           // global memory row
    float sumsq = 0.f;

    const float4* m4  = (const float4*)mem;
#if !HAVE_TDM
    const float4* dn4 = (const float4*)dn;
#endif

    // ---- prologue: stage tile 0 into buffer 0 ----
#if HAVE_TDM
    if (t < 32) tdm_dn_tile(dn, 0, dn_base);
#else
    {
        float4* dl4 = (float4*)dn_base;
#pragma unroll
        for (int j = 0; j < 32; ++j) {
            const int i4 = t + 256 * j;
            dl4[i4] = dn4[(long)(i4 >> 3) * (HH / 4) + (i4 & 7)];
        }
    }
#endif
    {
        float4 v0 = {0.f, 0.f, 0.f, 0.f}, v1 = {0.f, 0.f, 0.f, 0.f};
        if (gn < NN) {
            const long base = (long)gn * (HH / 4) + bq0;
            v0 = m4[base];
            v1 = m4[base + 1];
        }
        float4* bl4 = (float4*)b_base;
        bl4[brow * (KT / 4) + bq0]     = v0;
        bl4[brow * (KT / 4) + bq0 + 1] = v1;
        sumsq += v0.x * v0.x + v0.y * v0.y + v0.z * v0.z + v0.w * v0.w;
        sumsq += v1.x * v1.x + v1.y * v1.y + v1.z * v1.z + v1.w * v1.w;
    }
#if HAVE_TDM
    if (t < 32) __builtin_amdgcn_s_wait_tensorcnt(0);
#endif
    __syncthreads();

    // ---- main double-buffered K loop ----
    for (int kt = 0; kt < NTILES; ++kt) {
        const int cur = kt & 1;
        const int nxt = cur ^ 1;
        const bool hasnext = (kt + 1) < NTILES;

        // kick off staging for tile kt+1 (dn via TDM, B via VGPR loads)
        float4 nv0 = {0.f, 0.f, 0.f, 0.f}, nv1 = {0.f, 0.f, 0.f, 0.f};
        if (hasnext) {
#if HAVE_TDM
            if (t < 32) tdm_dn_tile(dn, kt + 1, dn_base + nxt * (BB * KT));
#else
            float4* dl4 = (float4*)(dn_base + nxt * (BB * KT));
#pragma unroll
            for (int j = 0; j < 32; ++j) {
                const int i4 = t + 256 * j;
                dl4[i4] = dn4[(long)(i4 >> 3) * (HH / 4)
                              + (kt + 1) * (KT / 4) + (i4 & 7)];
            }
#endif
            if (gn < NN) {
                const long base =
                    (long)gn * (HH / 4) + (kt + 1) * (KT / 4) + bq0;
                nv0 = m4[base];
                nv1 = m4[base + 1];
            }
        }

        // compute on current buffers: KT/4 WMMA K-steps, 16 M x 2 N tiles
        const float* dn_lds = dn_base + cur * (BB * KT);
        const float* b_lds  = b_base + cur * (NCHUNK * KT);
#pragma unroll
        for (int k4 = 0; k4 < KT / 4; ++k4) {
            const int koff = k4 * 4 + lg * 2;
            const v2f b0 = *(const v2f*)&b_lds[(g * 32 + ln) * KT + koff];
            const v2f b1 = *(const v2f*)&b_lds[(g * 32 + 16 + ln) * KT + koff];
#pragma unroll
            for (int m = 0; m < 16; ++m) {
                const v2f a =
                    *(const v2f*)&dn_lds[(mg * 256 + m * 16 + ln) * KT + koff];
                acc[m][0] = __builtin_amdgcn_wmma_f32_16x16x4_f32(
                    false, a, false, b0, (short)0, acc[m][0], false, false);
                acc[m][1] = __builtin_amdgcn_wmma_f32_16x16x4_f32(
                    false, a, false, b1, (short)0, acc[m][1], false, false);
            }
        }

        // complete B staging for next tile (stores land after compute)
        if (hasnext) {
            float4* bl4 = (float4*)(b_base + nxt * (NCHUNK * KT));
            bl4[brow * (KT / 4) + bq0]     = nv0;
            bl4[brow * (KT / 4) + bq0 + 1] = nv1;
            sumsq += nv0.x * nv0.x + nv0.y * nv0.y + nv0.z * nv0.z + nv0.w * nv0.w;
            sumsq += nv1.x * nv1.x + nv1.y * nv1.y + nv1.z * nv1.z + nv1.w * nv1.w;
        }
#if HAVE_TDM
        if (t < 32) __builtin_amdgcn_s_wait_tensorcnt(0);
#endif
        __syncthreads();
    }

    // finalize inverse norms (4 staging threads per memory row)
    sumsq += __shfl_xor(sumsq, 1, 4);
    sumsq += __shfl_xor(sumsq, 2, 4);
    if ((t & 3) == 0) invn[brow] = 1.0f / fmaxf(sqrtf(sumsq), EPSF);
    __syncthreads();

    // local argmax: C layout => lanes 0-15 hold N=ln / M=r, lanes 16-31 M=r+8
    const float s0f = invn[g * 32 + ln];
    const float s1f = invn[g * 32 + 16 + ln];
    const int n0 = nbase + g * 32 + ln;
    const int n1 = n0 + 16;
    const bool ok0 = n0 < NN;
    const bool ok1 = n1 < NN;
    float* lval = g ? lval1 : lval0;
    int*   lidx = g ? lidx1 : lidx0;

#pragma unroll
    for (int m = 0; m < 16; ++m) {
#pragma unroll
        for (int r = 0; r < 8; ++r) {
            const float v0 = ok0 ? acc[m][0][r] * s0f : -__builtin_inff();
            const float v1 = ok1 ? acc[m][1][r] * s1f : -__builtin_inff();
            float bv; int bi;
            if (v1 > v0) { bv = v1; bi = n1; } else { bv = v0; bi = n0; }
            // width-16 argmax reduction (first-index tie-break)
            for (int off = 8; off >= 1; off >>= 1) {
                const float ov = __shfl_xor(bv, off, 16);
                const int   oi = __shfl_xor(bi, off, 16);
                if (ov > bv || (ov == bv && oi < bi)) { bv = ov; bi = oi; }
            }
            if (ln == 0) {
                const int Ml = mg * 256 + m * 16 + r + lg * 8;
                lval[Ml] = bv;
                lidx[Ml] = bi;
            }
        }
    }
    __syncthreads();

    // combine the two N-groups, emit per-chunk partials
    for (int i = t; i < BB; i += 256) {
        float a0 = lval0[i]; int i0 = lidx0[i];
        const float a1 = lval1[i]; const int i1 = lidx1[i];
        if (a1 > a0 || (a1 == a0 && i1 < i0)) { a0 = a1; i0 = i1; }
        pval[(long)wg * BB + i] = a0;
        pidx[(long)wg * BB + i] = i0;
    }
}

// ---------------------------------------------------------------------------
// Kernel 3: reduce partials per row, gather labeled_logits, fuse epilogue
// ---------------------------------------------------------------------------
__global__ __launch_bounds__(256) void reduce_out_kernel(
    const float* __restrict__ pval, const int* __restrict__ pidx,
    const float* __restrict__ weak_logits, const float* __restrict__ labeled_logits,
    float* __restrict__ out, int nchunks)
{
    __shared__ float sval[256];
    __shared__ int   sidx[256];
    const int b = blockIdx.x;
    const int t = threadIdx.x;
    float bv = -__builtin_inff();
    int   bi = 0x7fffffff;
    for (int c = t; c < nchunks; c += 256) {
        const float v = pval[(long)c * BB + b];
        const int   i = pidx[(long)c * BB + b];
        if (v > bv || (v == bv && i < bi)) { bv = v; bi = i; }
    }
    sval[t] = bv; sidx[t] = bi;
    __syncthreads();
    for (int w = 128; w > 0; w >>= 1) {
        if (t < w) {
            const float ov = sval[t + w]; const int oi = sidx[t + w];
            if (ov > sval[t] || (ov == sval[t] && oi < sidx[t])) {
                sval[t] = ov; sidx[t] = oi;
            }
        }
        __syncthreads();
    }
    const int idx = sidx[0];
    for (int c = t; c < CC; c += 256) {
        out[(long)b * CC + c] =
            labeled_logits[(long)idx * CC + c] * 0.7f +
            (1.0f - 100000.0f) * weak_logits[(long)b * CC + c];
    }
}

// ---------------------------------------------------------------------------
extern "C" void kernel_launch(void* const* d_in, const int* in_sizes, int n_in,
                              void* d_out, int out_size, void* d_ws, size_t ws_size,
                              hipStream_t stream)
{
    const float* weak_data      = (const float*)d_in[0]; // [1024 x 768]
    const float* weak_logits    = (const float*)d_in[1]; // [1024 x 42]
    const float* labeled_memory = (const float*)d_in[2]; // [100000 x 768]
    const float* labeled_logits = (const float*)d_in[3]; // [100000 x 42]
    float* out = (float*)d_out;                          // [1024 x 42]

    const int nchunks = (NN + NCHUNK - 1) / NCHUNK;      // 1563

    // workspace layout
    float* dn   = (float*)d_ws;                          // BB*HH floats (3 MB)
    float* pval = dn + (size_t)BB * HH;                  // nchunks*BB floats
    int*   pidx = (int*)(pval + (size_t)nchunks * BB);   // nchunks*BB ints

    normalize_kernel<<<BB, 256, 0, stream>>>(weak_data, dn);

    const size_t lds_bytes =
        (size_t)(2 * BB * KT + 2 * NCHUNK * KT + NCHUNK) * sizeof(float) +
        (size_t)BB * 2 * (sizeof(float) + sizeof(int));  // ~288 KB (< 320 KB WGP)
    gemm_argmax_kernel<<<nchunks, 256, lds_bytes, stream>>>(
        dn, labeled_memory, pval, pidx);

    reduce_out_kernel<<<BB, 256, 0, stream>>>(
        pval, pidx, weak_logits, labeled_logits, out, nchunks);
}